// SemiMarkovConditionalRandomFieldWithTransition_26663156973970
// MI455X (gfx1250) — compile-verified
//
#include <hip/hip_runtime.h>
#include <hip/hip_bf16.h>

typedef __attribute__((ext_vector_type(2))) float v2f;
typedef __attribute__((ext_vector_type(8))) float v8f;

#define Bn 64
#define Sn 256
#define Wn 32
#define Tn 32
#define LDSP 36  // padded LDS row stride: 16B-aligned rows, conflict-free banks
#define TILE (Wn * LDSP)

#if __has_builtin(__builtin_amdgcn_global_load_async_to_lds_b128) && \
    __has_builtin(__builtin_amdgcn_s_wait_asynccnt)
#define HAVE_ASYNC_LDS 1
#else
#define HAVE_ASYNC_LDS 0
#endif

#if HAVE_ASYNC_LDS
// Exact pointee type the builtin expects: int __vector_size__(16)
typedef int v4i_gcc __attribute__((vector_size(16)));
typedef __attribute__((address_space(1))) v4i_gcc* gptr_v4i;  // global (AS1)
typedef __attribute__((address_space(3))) v4i_gcc* lptr_v4i;  // LDS (AS3)
// One b128 async DMA: copies 16 bytes from global address src+off to LDS address dst+off, tracked by ASYNCcnt.
// Generic->AS1/AS3 conversion via integer casts (the low 32 bits of the pointer are the LDS offset).
#define ASYNC_TILE_B128(srcp, dstp, off)                                  \
  __builtin_amdgcn_global_load_async_to_lds_b128(                         \
      (gptr_v4i)(unsigned long long)(srcp),                               \
      (lptr_v4i)(unsigned int)(unsigned long long)(dstp), (off), 0)
#endif

// ---------------------------------------------------------------------------
// init: lengths[b] = sum(mask[b,:])
// ---------------------------------------------------------------------------
__global__ __launch_bounds__(64) void crf_init_kernel(const int* __restrict__ mask,
                                                      int* __restrict__ lens) {
  int b = threadIdx.x;
  if (b < Bn) {
    int s = 0;
    for (int j = 0; j < Sn; ++j) s += mask[b * Sn + j];
    lens[b] = s;
  }
}

// ---------------------------------------------------------------------------
// Denominator: forward algorithm. One block (128 thr = 4 wave32) per batch.
// inner[d][t] = m_d + log( sum_{t'} exp(alpha_d[t']-m_d)*E[t',t] ) + emit[d][t]
// with E = exp(transitions); the 32x32x32 f32 matmul uses
// v_wmma_f32_16x16x4_f32 (2x2 C tiles, one per wave, 8 chained K=4 steps).
// Emit tiles are double-buffered in LDS and streamed with
// global_load_async_to_lds_b128 (ASYNCcnt) to overlap the serial chain.
// ---------------------------------------------------------------------------
__global__ __launch_bounds__(128) void crf_denominator_kernel(
    const float* __restrict__ logits,    // (B,S,W,T)
    const float* __restrict__ trans,     // (T,T)  trans[prev*T + cur]
    const float* __restrict__ start_tr,  // (T)
    const float* __restrict__ end_tr,    // (T)
    const int* __restrict__ lens,        // (B)
    float* __restrict__ out_den)         // (B)
{
  __shared__ __align__(16) float expA[TILE];     // slot s: exp(alpha_s - m_s)
  __shared__ __align__(16) float emitS[2 * TILE];// double-buffered emit tiles
  __shared__ __align__(16) float innerS[TILE];   // inner[d][t]
  __shared__ float mbuf[Wn];                     // per-slot row max

  const int b    = blockIdx.x;
  const int tid  = threadIdx.x;
  const int lane = tid & 31;
  const int wv   = tid >> 5;
  const int len  = lens[b];

  for (int i = tid; i < TILE; i += 128) expA[i] = 0.0f;
  if (tid < Wn) mbuf[tid] = 0.0f;

  // each wave owns one 16x16 C tile of the 32x32 result
  const int mbase   = (wv >> 1) << 4;
  const int nbase   = (wv & 1) << 4;
  const int halfsel = lane >> 4;   // 0: lanes 0-15, 1: lanes 16-31
  const int l15     = lane & 15;

  // B operand tile: E = exp(transitions), in v_wmma_f32_16x16x4_f32 B layout:
  //   element i of chunk c -> B[k][n], k = 4c + 2*halfsel + i, n = nbase + l15
  v2f Breg[8];
#pragma unroll
  for (int c2 = 0; c2 < 8; ++c2) {
    const int k0 = 4 * c2 + 2 * halfsel;
    const int n  = nbase + l15;
    Breg[c2].x = __expf(trans[k0 * Tn + n]);
    Breg[c2].y = __expf(trans[(k0 + 1) * Tn + n]);
  }

  const float stv = start_tr[lane];  // start_tr[t], t = lane (used by wave 0)
  const float env = end_tr[lane];

  const float* lgB = logits + (size_t)b * Sn * Wn * Tn;
  // this thread's 8-element chunk of each 32x32 emit tile
  const int erow = (tid * 8) >> 5;     // tile row
  const int ecol = (tid * 8) & 31;     // tile col (0,8,16,24)
  const int eoff = erow * LDSP + ecol; // padded LDS offset (16B aligned)

#if HAVE_ASYNC_LDS
  {  // prologue: kick off tile 0 into buffer 0
    const float* src = lgB + tid * 8;
    float* dst = &emitS[eoff];
    ASYNC_TILE_B128(src, dst, 0);
    ASYNC_TILE_B128(src, dst, 16);
  }
#endif
  __syncthreads();

  for (int j = 0; j < Sn; ++j) {
    const bool active = (j < len);
    const int slot = j & 31;
    const int prev = (j - 1) & 31;
    float* emitC = &emitS[(j & 1) * TILE];

#if HAVE_ASYNC_LDS
    // stream tile j+1 into the other buffer; wait until tile j is resident
    if (j + 1 < Sn) {
      const float* src = lgB + (size_t)(j + 1) * (Wn * Tn) + tid * 8;
      float* dst = &emitS[((j + 1) & 1) * TILE + eoff];
      ASYNC_TILE_B128(src, dst, 0);
      ASYNC_TILE_B128(src, dst, 16);
      __builtin_amdgcn_s_wait_asynccnt(2);  // in-order: tile j complete
    } else {
      __builtin_amdgcn_s_wait_asynccnt(0);
    }
#else
    {  // synchronous fallback: vectorized b128 copy of tile j
      const float4* src =
          (const float4*)(lgB + (size_t)j * (Wn * Tn) + tid * 8);
      float4* dst = (float4*)&emitC[eoff];
      dst[0] = src[0];
      dst[1] = src[1];
      if (j + 1 < Sn)
        __builtin_prefetch(lgB + (size_t)(j + 1) * (Wn * Tn) + tid * 8, 0, 1);
    }
#endif
    __syncthreads();

    // ---- WMMA phase: S[d][t] = sum_{t'} expA[slot(d)][t'] * E[t'][t] -------
    if (active) {
      v8f acc = {};
      // A layout (16x4 f32): element i of chunk c -> A[m][k],
      //   m = mbase + l15,  k = 4c + 2*halfsel + i
      const int mrow  = mbase + l15;
      const int aslot = (j - 1 - mrow) & 31;  // circular slot of alpha_{j-1-d}
      const float* arow = &expA[aslot * LDSP];
#pragma unroll
      for (int c2 = 0; c2 < 8; ++c2) {
        const int k0 = 4 * c2 + 2 * halfsel;
        v2f a;
        a.x = arow[k0];
        a.y = arow[k0 + 1];
        acc = __builtin_amdgcn_wmma_f32_16x16x4_f32(
            /*neg_a=*/false, a, /*neg_b=*/false, Breg[c2],
            /*c_mod=*/(short)0, acc, /*reuse_a=*/false, /*reuse_b=*/false);
      }
      // D layout: vgpr v -> m = mbase + 8*halfsel + v, n = nbase + l15
#pragma unroll
      for (int v = 0; v < 8; ++v) {
        const int d   = mbase + 8 * halfsel + v;
        const int col = nbase + l15;
        const float mv = mbuf[(j - 1 - d) & 31];
        innerS[d * LDSP + col] = __logf(acc[v]) + mv + emitC[d * LDSP + col];
      }
    }
    __syncthreads();

    // ---- reduce over span d (wave 0; t = lane), store new alpha slot -------
    if (wv == 0) {
      if (active) {
        const int t = lane;
        const int dmax = (j < Wn) ? j : (Wn - 1);
        float mx = -1e30f;
        for (int d = 0; d <= dmax; ++d) {
          const float v = (d == j) ? (emitC[j * LDSP + t] + stv)
                                   : innerS[d * LDSP + t];
          mx = fmaxf(mx, v);
        }
        float se = 0.0f;
        for (int d = 0; d <= dmax; ++d) {
          const float v = (d == j) ? (emitC[j * LDSP + t] + stv)
                                   : innerS[d * LDSP + t];
          se += __expf(v - mx);
        }
        const float an = mx + __logf(se);  // alpha_j[t]
        float rm = an;                     // row max over the 32 tags
#pragma unroll
        for (int off = 16; off > 0; off >>= 1)
          rm = fmaxf(rm, __shfl_xor(rm, off, 32));
        expA[slot * LDSP + t] = __expf(an - rm);
        if (lane == 0) mbuf[slot] = rm;
      } else {  // masked step: alpha_j = alpha_{j-1}
        expA[slot * LDSP + lane] = expA[prev * LDSP + lane];
        if (lane == 0) mbuf[slot] = mbuf[prev];
      }
    }
    __syncthreads();
  }

  // ---- log_den[b] = logsumexp_t(alpha_last[t] + end_tr[t]) -----------------
  if (wv == 0) {
    const int slot = (Sn - 1) & 31;
    float p = expA[slot * LDSP + lane] * __expf(env);
#pragma unroll
    for (int off = 16; off > 0; off >>= 1)
      p += __shfl_xor(p, off, 32);
    if (lane == 0) out_den[b] = mbuf[slot] + __logf(p);
  }
}

// ---------------------------------------------------------------------------
// Numerator: one block per (chunk, b); 256 threads, one (j,d) pair each.
// Deterministic: per-chunk partials, no float atomics.
// ---------------------------------------------------------------------------
__global__ __launch_bounds__(256) void crf_numerator_kernel(
    const float* __restrict__ logits,   // (B,S,W,T)
    const int* __restrict__ tags,       // (B,S,W)
    const int* __restrict__ mask,       // (B,S)
    const float* __restrict__ trans,    // (T,T)
    const float* __restrict__ start_tr, // (T)
    const float* __restrict__ end_tr,   // (T)
    const int* __restrict__ lens,       // (B)
    float* __restrict__ part)           // (B, 32) chunk partial sums
{
  __shared__ float transS[Tn * Tn];
  __shared__ float red[256];

  const int b     = blockIdx.y;
  const int chunk = blockIdx.x;
  const int tid   = threadIdx.x;

  for (int i = tid; i < Tn * Tn; i += 256) transS[i] = trans[i];
  __syncthreads();

  const int p = chunk * 256 + tid;  // 0..8191
  const int j = p >> 5;
  const int d = p & 31;

  float score = 0.0f;
  const int tg  = tags[((size_t)b * Sn + j) * Wn + d];
  const float tm = (float)mask[b * Sn + j];
  if (d <= j) {
    float emitc = 0.0f;
    if (tg != 0)  // DEFAULT_TAG == 0
      emitc = logits[(((size_t)b * Sn + j) * Wn + d) * Tn + tg] * tm;
    float tr;
    if (d == j) {
      tr = start_tr[tg];
    } else {
      tr = 0.0f;
      const int* prow = tags + ((size_t)b * Sn + (j - d - 1)) * Wn;
#pragma unroll 4
      for (int w = 0; w < Wn; ++w) {
        const int pt = prow[w];
        if (pt != 0) tr += transS[pt * Tn + tg];
      }
    }
    score = tr * tm + emitc;
  }
  // end-transition contribution (once per batch, folded into chunk 0)
  if (chunk == 0 && tid < Wn) {
    const int len = lens[b];
    const int lt = tags[((size_t)b * Sn + (len - 1)) * Wn + tid];
    if (lt != 0) score += end_tr[lt];
  }

  red[tid] = score;
  __syncthreads();
  for (int s = 128; s > 0; s >>= 1) {
    if (tid < s) red[tid] += red[tid + s];
    __syncthreads();
  }
  if (tid == 0) part[b * 32 + chunk] = red[0];
}

// ---------------------------------------------------------------------------
// Finalize: log_num[b] = sum(part[b,:]) (fixed order), loss = mean(num - den)
// d_out[0] = batch_loss, d_out[1..64] = log_num
// ---------------------------------------------------------------------------
__global__ __launch_bounds__(64) void crf_finalize_kernel(
    const float* __restrict__ part, const float* __restrict__ den,
    float* __restrict__ out) {
  __shared__ float red[64];
  const int b = threadIdx.x;
  float num = 0.0f;
  for (int c = 0; c < 32; ++c) num += part[b * 32 + c];
  out[1 + b] = num;
  red[b] = num - den[b];
  __syncthreads();
  for (int s = 32; s > 0; s >>= 1) {
    if (b < s) red[b] += red[b + s];
    __syncthreads();
  }
  if (b == 0) out[0] = red[0] / (float)Bn;
}

// ---------------------------------------------------------------------------
extern "C" void kernel_launch(void* const* d_in, const int* in_sizes, int n_in,
                              void* d_out, int out_size, void* d_ws, size_t ws_size,
                              hipStream_t stream) {
  const float* logits  = (const float*)d_in[0];  // (B,S,W,T) f32
  const int* tags      = (const int*)d_in[1];    // (B,S,W) i32
  const int* mask      = (const int*)d_in[2];    // (B,S) i32
  const float* trans   = (const float*)d_in[3];  // (T,T)
  const float* s_tr    = (const float*)d_in[4];  // (T)
  const float* e_tr    = (const float*)d_in[5];  // (T)
  float* out = (float*)d_out;                    // [0]=loss, [1..64]=log_num

  char* ws   = (char*)d_ws;
  float* part = (float*)ws;              // B*32 f32  = 8192 B
  float* den  = (float*)(ws + 8192);     // B f32     = 256 B
  int*   lens = (int*)(ws + 8448);       // B i32     = 256 B

  crf_init_kernel<<<1, 64, 0, stream>>>(mask, lens);
  crf_numerator_kernel<<<dim3(32, Bn), 256, 0, stream>>>(logits, tags, mask,
                                                         trans, s_tr, e_tr,
                                                         lens, part);
  crf_denominator_kernel<<<Bn, 128, 0, stream>>>(logits, trans, s_tr, e_tr,
                                                 lens, den);
  crf_finalize_kernel<<<1, 64, 0, stream>>>(part, den, out);
}